// DPAAUser2D_26001732010121
// MI455X (gfx1250) — compile-verified
//
#include <hip/hip_runtime.h>
#include <hip/hip_bf16.h>

// Problem constants from the reference
#define B_  8
#define C_  19
#define H_  512
#define W_  512
// adjusted: 64x64 per (b,c); patches: [B,C,16,256]; attn: [B,C,1024,16]

typedef __attribute__((ext_vector_type(2))) float v2f;
typedef __attribute__((ext_vector_type(8))) float v8f;

// ---------------------------------------------------------------------------
// Kernel A: per-pixel channel argmax -> 8x8 pooled counts/64 -> unfolded
// patches [B,C,16,256].  One block per (b, ah) where ah is the 8-pixel-high
// block row (64 of them).  512 threads = one per column; each thread owns the
// 8 rows of its column -> fully coalesced 2KB row reads per channel.
// ---------------------------------------------------------------------------
__global__ __launch_bounds__(512) void dpaa_pool_argmax(
    const float* __restrict__ x, float* __restrict__ patches) {
  const int b  = blockIdx.x >> 6;   // 0..7
  const int ah = blockIdx.x & 63;   // 0..63  (adjusted row)
  const int w  = threadIdx.x;       // 0..511 (pixel column)

  const float* xb = x + (((size_t)b * C_) * H_ + (size_t)ah * 8) * W_ + w;

  float bestv[8];
  int   bestc[8];
#pragma unroll
  for (int r = 0; r < 8; ++r) { bestv[r] = xb[r * W_]; bestc[r] = 0; }
  for (int c = 1; c < C_; ++c) {
    const float* xc = xb + (size_t)c * H_ * W_;
#pragma unroll
    for (int r = 0; r < 8; ++r) {
      float v = xc[r * W_];
      if (v > bestv[r]) { bestv[r] = v; bestc[r] = c; }  // first-max tiebreak
    }
  }

  __shared__ int counts[64 * C_];   // [aw][c]
  for (int i = threadIdx.x; i < 64 * C_; i += 512) counts[i] = 0;
  __syncthreads();

  const int awB = w >> 3;
#pragma unroll
  for (int r = 0; r < 8; ++r) atomicAdd(&counts[awB * C_ + bestc[r]], 1);
  __syncthreads();

  for (int i = threadIdx.x; i < 64 * C_; i += 512) {
    int aw = i / C_;
    int c  = i % C_;
    // unfold: l = (ah>>4)*4 + (aw>>4); kidx = (ah&15)*16 + (aw&15)
    int l    = ((ah >> 4) << 2) + (aw >> 4);
    int kidx = ((ah & 15) << 4) + (aw & 15);
    patches[((((size_t)b * C_ + c) * 16) + l) * 256 + kidx] =
        (float)counts[i] * 0.015625f;  // /64
  }
}

// ---------------------------------------------------------------------------
// Kernel B: per-(b,c) GEMM [1024x16]x[16x256] via chained
// v_wmma_f32_16x16x4_f32, fused fold + elementwise epilogue.
// One wave per 16x16 output tile; 4 waves per block.
// ---------------------------------------------------------------------------
__global__ __launch_bounds__(128) void dpaa_wmma_fold(
    const float* __restrict__ attn, const float* __restrict__ patches,
    const float* __restrict__ x, float* __restrict__ out0,
    float* __restrict__ out1) {
  const int wave = threadIdx.x >> 5;
  const int lane = threadIdx.x & 31;
  const int t    = blockIdx.x * 4 + wave;      // global tile id
  const int nt   = t & 15;                     // N tile (0..15)
  const int mt   = (t >> 4) & 63;              // M tile (0..63)
  const int bc   = t >> 10;                    // fused (b*C + c), 0..151

  const int half  = lane >> 4;   // 0: K even pair, 1: K odd pair (per ISA A/B layout)
  const int mrow  = lane & 15;   // A row within tile
  const int nlane = lane & 15;   // B/D column within tile

  // A: attn row is 16 contiguous floats
  const float* arow  = attn + (((size_t)bc * 1024) + (size_t)mt * 16 + mrow) * 16;
  // B: patches[bc][l=K][n], row stride 256
  const float* pbase = patches + (size_t)bc * 4096 + nt * 16 + nlane;

  v8f acc = {};
#pragma unroll
  for (int kk = 0; kk < 4; ++kk) {             // K = 16 = 4 steps of K=4
    const int kb = kk * 4;
    v2f a, bf;
    // A 16x4 layout: VGPR0 = K{0|2}, VGPR1 = K{1|3} across lane halves
    a.x  = arow[kb + 2 * half + 0];
    a.y  = arow[kb + 2 * half + 1];
    // B 4x16 layout mirrors A: row striped across lanes within each VGPR
    bf.x = pbase[(size_t)(kb + 2 * half + 0) * 256];
    bf.y = pbase[(size_t)(kb + 2 * half + 1) * 256];
    acc = __builtin_amdgcn_wmma_f32_16x16x4_f32(
        /*neg_a=*/false, a, /*neg_b=*/false, bf,
        /*c_mod=*/(short)0, acc, /*reuse_a=*/false, /*reuse_b=*/false);
  }

  // Epilogue: fold + y2 = x*(1+corr)^2, out1 = corr
  const size_t plane = (size_t)bc * (size_t)(H_ * W_);
#pragma unroll
  for (int r = 0; r < 8; ++r) {
    // D layout: VGPR r -> M = r (lanes 0-15) or r+8 (lanes 16-31)
    const int m  = mt * 16 + r + 8 * half;     // modulated row, 0..1023
    const int hh = (m >> 5) * 16 + nt;         // k>>4 == nt
    const int ww = (m & 31) * 16 + nlane;      // k&15 == nlane
    const size_t idx = plane + (size_t)hh * W_ + ww;
    const float corr = acc[r];
    const float xv   = x[idx];
    const float tt   = 1.0f + corr;
    out0[idx] = xv * tt * tt;                  // (x + c*x) + c*(x + c*x)
    out1[idx] = corr;
  }
}

// ---------------------------------------------------------------------------
extern "C" void kernel_launch(void* const* d_in, const int* in_sizes, int n_in,
                              void* d_out, int out_size, void* d_ws, size_t ws_size,
                              hipStream_t stream) {
  const float* x    = (const float*)d_in[0];  // [8,19,512,512]
  const float* attn = (const float*)d_in[1];  // [8,19,1024,16]
  float* out0 = (float*)d_out;
  float* out1 = out0 + (size_t)B_ * C_ * H_ * W_;
  float* patches = (float*)d_ws;              // [8,19,16,256] = 2.49 MB

  dpaa_pool_argmax<<<B_ * 64, 512, 0, stream>>>(x, patches);

  const int tiles = B_ * C_ * 64 * 16;        // 155648, divisible by 4
  dpaa_wmma_fold<<<tiles / 4, 128, 0, stream>>>(attn, patches, x, out0, out1);
}